// QRMSNorm_25615184953512
// MI455X (gfx1250) — compile-verified
//
#include <hip/hip_runtime.h>

typedef _Float16 half_t;
typedef __attribute__((ext_vector_type(8)))  _Float16     v8h;
typedef __attribute__((ext_vector_type(4)))  float        v4f;
typedef __attribute__((ext_vector_type(4)))  unsigned int u32x4;
typedef __attribute__((ext_vector_type(8)))  int          i32x8;

#define HDIM 4096
#define TPB  256            // 8 waves (wave32)
#define EPT  (HDIM / TPB)   // 16 fp16 elements per thread
#define RPB  4              // rows per block (amortize weight staging)
#define NW   (TPB / 32)

__global__ __launch_bounds__(TPB) void qrmsnorm_dynq_kernel(
    const half_t* __restrict__ x,
    const half_t* __restrict__ w,
    float* __restrict__ out,
    int rows)
{
  __shared__ half_t lds_w[HDIM];          // 8 KB weight tile staged by TDM
  __shared__ float  s_red[2][NW];

  const int tid  = threadIdx.x;
  const int wave = tid >> 5;
  const int lane = tid & 31;

  // ---- Stage the shared weight row into LDS via the Tensor Data Mover ----
#if defined(__gfx1250__)
  if (tid < 32) {   // one wave issues the TDM descriptor (wave-uniform branch)
    // Generic LDS address: low 32 bits == workgroup-relative LDS byte offset.
    unsigned int lds_off = (unsigned int)(unsigned long long)&lds_w[0];
    unsigned long long ga = (unsigned long long)w;
    // D# group 0 (128b): count=1, lds_addr, global_addr[56:0], type=2
    u32x4 g0;
    g0.x = 1u;
    g0.y = lds_off;
    g0.z = (unsigned int)ga;
    g0.w = (unsigned int)((ga >> 32) & 0x01FFFFFFull) | (2u << 30);
    // D# group 1 (256b): data_size=2B, 1-D tensor: dim0=tile_dim0=4096
    i32x8 g1;
    g1[0] = (int)(1u << 16);                                 // wg_mask=0, data_size=2B
    g1[1] = (int)((unsigned)(HDIM & 0xFFFF) << 16);          // tensor_dim0[15:0]
    g1[2] = (int)(((HDIM >> 16) & 0xFFFF) | (1u << 16));     // dim0[31:16], tensor_dim1=1
    g1[3] = (int)((unsigned)(HDIM & 0xFFFF) << 16);          // tile_dim0=4096
    g1[4] = 0;                                               // tile_dim1=0 (1D), tile_dim2=0
    g1[5] = HDIM;                                            // tensor_dim0_stride
    g1[6] = 0;
    g1[7] = 0;
    // 2-group form: VADDR2/VADDR3 = NULL (tensors up to 2D), per ISA §7.3
    asm volatile("tensor_load_to_lds %0, %1"
                 :
                 : "s"(g0), "s"(g1)
                 : "memory");
  }
  __builtin_amdgcn_s_wait_tensorcnt(0);   // no-op for waves that didn't issue
#else
  // Fallback (host pass / non-gfx1250): cooperative 128-bit copy
  for (int i = tid; i < HDIM / 8; i += TPB)
    ((v8h*)lds_w)[i] = ((const v8h*)w)[i];
#endif
  __syncthreads();

  // Hoist this thread's 16 weights to f32 registers (reused across RPB rows)
  float wf[EPT];
  {
    const v8h* wl = (const v8h*)&lds_w[tid * EPT];
    v8h a = wl[0], b = wl[1];
#pragma unroll
    for (int i = 0; i < 8; ++i) { wf[i] = (float)a[i]; wf[8 + i] = (float)b[i]; }
  }

  for (int r = 0; r < RPB; ++r) {
    const int row = blockIdx.x * RPB + r;
    if (row >= rows) break;

    // 2x 128-bit loads of this thread's 16 fp16 values
    const v8h* xp = (const v8h*)(x + (size_t)row * HDIM + tid * EPT);
    v8h a = xp[0], b = xp[1];
    float xf[EPT];
#pragma unroll
    for (int i = 0; i < 8; ++i) { xf[i] = (float)a[i]; xf[8 + i] = (float)b[i]; }

    // ---- sum of squares (fp32 accumulation) ----
    float ss = 0.0f;
#pragma unroll
    for (int i = 0; i < EPT; ++i) ss = __builtin_fmaf(xf[i], xf[i], ss);
#pragma unroll
    for (int m = 16; m >= 1; m >>= 1) ss += __shfl_xor(ss, m, 32);   // wave32
    if (lane == 0) s_red[0][wave] = ss;
    __syncthreads();
    float var = 0.0f;
#pragma unroll
    for (int iw = 0; iw < NW; ++iw) var += s_red[0][iw];             // LDS broadcast
    const float rs = rsqrtf(var * (1.0f / (float)HDIM) + 1e-6f);

    // ---- normalize + abs-max ----
    float xn[EPT];
    float amax = 0.0f;
#pragma unroll
    for (int i = 0; i < EPT; ++i) {
      xn[i] = xf[i] * wf[i] * rs;
      amax  = fmaxf(amax, fabsf(xn[i]));
    }
#pragma unroll
    for (int m = 16; m >= 1; m >>= 1) amax = fmaxf(amax, __shfl_xor(amax, m, 32));
    if (lane == 0) s_red[1][wave] = amax;
    __syncthreads();
    float gmax = 0.0f;
#pragma unroll
    for (int iw = 0; iw < NW; ++iw) gmax = fmaxf(gmax, s_red[1][iw]);

    // ---- quantize (RNE round, clip to [-128,127]); store as f32 ----
    const float inv = gmax > 0.0f ? 127.0f / gmax : 0.0f;
    v4f o[EPT / 4];
#pragma unroll
    for (int i = 0; i < EPT; ++i) {
      float q = rintf(xn[i] * inv);
      q = fminf(fmaxf(q, -128.0f), 127.0f);
      o[i >> 2][i & 3] = q;
    }
    v4f* op = (v4f*)(out + (size_t)row * HDIM + tid * EPT);
#pragma unroll
    for (int i = 0; i < EPT / 4; ++i) op[i] = o[i];                  // 4x B128 stores
    if (tid == 0) out[(size_t)rows * HDIM + row] = gmax * (1.0f / 127.0f);
  }
}

extern "C" void kernel_launch(void* const* d_in, const int* in_sizes, int n_in,
                              void* d_out, int out_size, void* d_ws, size_t ws_size,
                              hipStream_t stream) {
  const half_t* x = (const half_t*)d_in[0];   // hidden_states fp16 [B,S,H]
  const half_t* w = (const half_t*)d_in[1];   // weight fp16 [H]
  float* out = (float*)d_out;                 // [B*S*H] q (as f32) then [B*S] scales
  const int rows = in_sizes[0] / HDIM;        // B*S tokens
  const int grid = (rows + RPB - 1) / RPB;
  qrmsnorm_dynq_kernel<<<grid, TPB, 0, stream>>>(x, w, out, rows);
}